// BumpX_49143015801560
// MI455X (gfx1250) — compile-verified
//
#include <hip/hip_runtime.h>

// MI455X / gfx1250 kernel for the "BumpX" banded-softmask weighted average.
//
// Reasoning (compile-only, argued from HW specs):
//  * Bytes moved: ~1.5 MB total @ 23.3 TB/s -> memory is irrelevant.
//  * Cost is the transcendental mask eval (4 exp + 2 log + 3 rcp per element).
//  * mask[f,g] underflows to EXACTLY +0.0f for |g-f| >= ~10 in fp32 (softplus
//    clamp at 1e-6 -> exp(-1e6) == 0, gg = f1/f1 == 1.0 exactly), so a banded
//    kernel (margin +-32) is bit-equivalent to the dense reference sum and cuts
//    transcendental work ~13x.
//  * The einsum + row-sum are fused into V_WMMA_F32_16X16X4_F32 tiles:
//    A = 16x4 fp32 mask tile (computed in-register, no conversion loss),
//    B = 4x16 with col0 = x[g..g+3], col1 = ones  =>
//    C col0 = weighted sum, C col1 = denominator, accumulated in fp32.
//  * B is built branch-free (loop-invariant lane selectors + FMA) so the hot
//    loop has no EXEC manipulation: EXEC stays all-ones for WMMA, and the
//    per-iteration SALU/branch overhead from round 1 is gone.

typedef __attribute__((ext_vector_type(2))) float v2f;
typedef __attribute__((ext_vector_type(8))) float v8f;

#define FDIM    1024
#define BAND    32
#define ROWS_PB 64      // rows per block = 4 waves * 16-row strips
#define THREADS 128

__device__ __forceinline__ float fast_softplus(float t) {
    // log(1 + exp(t)) = max(t, 0) + log(1 + exp(-|t|)); exp arg <= 0 -> no overflow
    float e = __expf(-fabsf(t));
    return fmaxf(t, 0.0f) + __logf(1.0f + e);
}

__device__ __forceinline__ float bump_mask(float arg) {
    float sp1 = fmaxf(fast_softplus(arg),        1e-6f);
    float sp2 = fmaxf(fast_softplus(1.0f - arg), 1e-6f);
    float f1  = __expf(-__builtin_amdgcn_rcpf(sp1));
    float f2  = __expf(-__builtin_amdgcn_rcpf(sp2));
    // mask = 1 - f1/(f1+f2) = f2/(f1+f2); exactly 0 when f2 underflows
    return f2 * __builtin_amdgcn_rcpf(f1 + f2);
}

__global__ __launch_bounds__(THREADS)
void bumpx_kernel(const float* __restrict__ x,
                  const float* __restrict__ aa,
                  float* __restrict__ out) {
    __shared__ float xs[FDIM];
    __shared__ float lw[ROWS_PB];   // weighted sums  (C column 0)
    __shared__ float ldn[ROWS_PB];  // denominators   (C column 1)

    const int bl   = blockIdx.y;           // (b,l) pair
    const int tid  = threadIdx.x;

    const float* xrow = x  + (size_t)bl * FDIM;
    const float* arow = aa + (size_t)bl * FDIM;

    // Stage x[b,l,:] in LDS (feeds B-matrix construction each chunk)
    for (int i = tid; i < FDIM; i += THREADS) xs[i] = xrow[i];
    __syncthreads();

    const int wid  = tid >> 5;             // wave id: 16-row strip per wave
    const int lane = tid & 31;
    const int n    = lane & 15;
    const int half = lane >> 4;

    const int fbase = blockIdx.x * ROWS_PB + wid * 16;
    const int frow  = fbase + n;           // this lane's matrix row

    // Per-row constants: arg = d^2 * inv_den - c0
    const float a       = arow[frow];
    const float inv_den = __builtin_amdgcn_rcpf(fmaf(6.0f, a, 9.0f));
    const float c0      = a * a * inv_den;

    // Loop-invariant B-column selectors: col 0 = x, col 1 = ones, rest 0
    const float sel_x = (n == 0) ? 1.0f : 0.0f;
    const float sel_1 = (n == 1) ? 1.0f : 0.0f;

    // Band limits (wave-uniform): mask is exactly 0 outside |g-f| >= ~10
    int gstart = fbase - BAND;        if (gstart < 0)    gstart = 0;
    int gend   = fbase + 16 + BAND;   if (gend   > FDIM) gend   = FDIM;

    v8f acc = {};  // 16x16 fp32 accumulator: col0 = sum(mask*x), col1 = sum(mask)

    for (int g0 = gstart; g0 < gend; g0 += 4) {
        // A tile (16x4 fp32): lane holds row n, K = {2*half, 2*half+1}
        // (ISA layout: VGPR0 = K0 / K2, VGPR1 = K1 / K3 across lane halves)
        const int gA  = g0 + 2 * half;
        const float d0 = (float)(gA     - frow);
        const float d1 = (float)(gA + 1 - frow);
        v2f A;
        A.x = bump_mask(fmaf(d0 * d0, inv_den, -c0));
        A.y = bump_mask(fmaf(d1 * d1, inv_den, -c0));

        // B tile (4x16 fp32): VGPR0 = row K0 (lanes 0-15) / K2 (16-31),
        //                     VGPR1 = row K1 / K3; column n per lane.
        // Branch-free: all lanes load (ds_load_b64, gA is even), then select.
        const float xv0 = xs[gA];
        const float xv1 = xs[gA + 1];
        v2f B;
        B.x = fmaf(sel_x, xv0, sel_1);
        B.y = fmaf(sel_x, xv1, sel_1);

        // EXEC is all-ones here (no divergent control flow in the loop; bounds
        // are wave-uniform). Compiler inserts WMMA hazard NOPs.
        acc = __builtin_amdgcn_wmma_f32_16x16x4_f32(false, A, false, B,
                                                    (short)0, acc, false, false);
    }

    // C layout: VGPR j -> (M=j, N=lane) for lanes 0-15, (M=8+j, N=lane-16) else.
    // Column 0 lives in lanes {0,16}, column 1 in lanes {1,17}.
    if (n < 2) {
        float* dst = (n == 0) ? lw : ldn;
        #pragma unroll
        for (int j = 0; j < 8; ++j) dst[wid * 16 + half * 8 + j] = acc[j];
    }
    // Same-wave LDS ops are in-order: stores above are visible to loads below.
    if (lane < 16) {
        float w = lw[wid * 16 + lane];
        float d = ldn[wid * 16 + lane];
        out[(size_t)bl * FDIM + fbase + lane] = w / d;  // precise div, 131K total
    }
}

extern "C" void kernel_launch(void* const* d_in, const int* in_sizes, int n_in,
                              void* d_out, int out_size, void* d_ws, size_t ws_size,
                              hipStream_t stream) {
    const float* x  = (const float*)d_in[0];
    const float* aa = (const float*)d_in[1];
    float* out      = (float*)d_out;

    const int BL = in_sizes[0] / FDIM;       // B*L = 128 for the reference shapes
    dim3 grid(FDIM / ROWS_PB, BL);           // 16 x 128 = 2048 workgroups
    bumpx_kernel<<<grid, THREADS, 0, stream>>>(x, aa, out);
}